// RPN_61924838473796
// MI455X (gfx1250) — compile-verified
//
#include <hip/hip_runtime.h>
#include <stdint.h>

// ---------------- problem constants ----------------
#define NBATCH   16
#define KTOT     4768          // 1000+1000+1000+1000+768 selected per image
#define NEGV     (-1e9f)
#define BBOX_CLIPV 4.135166556742356f   // log(1000/16)
#define IMGSZ    1024.0f
#define NMS_THR  0.7f

constexpr int LH[5]    = {256, 128, 64, 32, 16};
constexpr int LW[5]    = {256, 128, 64, 32, 16};
constexpr int LBASE[6] = {0, 196608, 245760, 258048, 261120, 261888};
constexpr int LK[5]    = {1000, 1000, 1000, 1000, 768};
constexpr int LKOFF[5] = {0, 1000, 2000, 3000, 4000};

// ---------------- CDNA5 TDM helpers ----------------
typedef unsigned int uv4 __attribute__((ext_vector_type(4)));
typedef int          iv4 __attribute__((ext_vector_type(4)));
typedef int          iv8 __attribute__((ext_vector_type(8)));

// 1-D (row) tensor DMA: global -> LDS, nElem 4-byte elements.
// D# group0: count=1, lds_addr, 57-bit global addr, type=2 ("image").
// D# group1: data_size=4B, tensor_dim0=nElem, tensor_dim1=1,
//            tile_dim0=nElem (<=65535), tile_dim1=1, stride0=nElem.
__device__ __forceinline__ void tdm_load_1d(unsigned ldsOff, const void* gptr,
                                            unsigned nElem) {
#if defined(__HIP_DEVICE_COMPILE__) && __HIP_DEVICE_COMPILE__
  unsigned long long ga = (unsigned long long)(uintptr_t)gptr;
  uv4 g0;
  g0.x = 1u;                                            // count=1, user mode
  g0.y = ldsOff;                                        // lds_addr (bytes)
  g0.z = (unsigned)(ga & 0xFFFFFFFFull);                // global_addr lo
  g0.w = (unsigned)((ga >> 32) & 0x01FFFFFFull)         // global_addr hi
       | (2u << 30);                                    // type=2
  iv8 g1;
  g1[0] = (int)(2u << 16);                              // data_size=4B
  g1[1] = (int)((nElem & 0xFFFFu) << 16);               // tensor_dim0 lo16
  g1[2] = (int)((nElem >> 16) | (1u << 16));            // dim0 hi | dim1 lo16=1
  g1[3] = (int)((nElem & 0xFFFFu) << 16);               // dim1 hi=0 | tile_dim0
  g1[4] = 1;                                            // tile_dim1=1, tile_dim2=0
  g1[5] = (int)nElem;                                   // tensor_dim0_stride lo
  g1[6] = 0;
  g1[7] = 0;
  iv4 z4 = {0, 0, 0, 0};
  iv8 z8 = {0, 0, 0, 0, 0, 0, 0, 0};
  // This toolchain declares the 6-arg form (probe: "expected 6").
  __builtin_amdgcn_tensor_load_to_lds(g0, g1, z4, z4, z8, 0);
#else
  (void)ldsOff; (void)gptr; (void)nElem;
#endif
}

__device__ __forceinline__ void tdm_wait0() {
#if defined(__HIP_DEVICE_COMPILE__) && __HIP_DEVICE_COMPILE__
  __builtin_amdgcn_s_wait_tensorcnt(0);
#endif
}

// ---------------- misc helpers ----------------
__device__ __forceinline__ unsigned fkey(float f) {
  unsigned u = __float_as_uint(f);
  return (u & 0x80000000u) ? ~u : (u | 0x80000000u);   // order-preserving
}

// ---------------- kernel 0: init workspace ----------------
__global__ void rpn_init(int* __restrict__ selIdx, float* __restrict__ selScr) {
  int g = blockIdx.x * blockDim.x + threadIdx.x;
  if (g < NBATCH * KTOT) { selIdx[g] = 0; selScr[g] = NEGV; }
}

// ---------------- kernel 1: per-(batch,level) exact top-k --------------
// MSB radix select over order-preserving uint keys; 4 histogram passes.
__global__ __launch_bounds__(1024) void rpn_topk(
    const float* __restrict__ c0, const float* __restrict__ c1,
    const float* __restrict__ c2, const float* __restrict__ c3,
    const float* __restrict__ c4,
    int* __restrict__ selIdx, float* __restrict__ selScr) {
  const int b = blockIdx.x / 5;
  const int l = blockIdx.x % 5;
  const float* cls;
  switch (l) { case 0: cls = c0; break; case 1: cls = c1; break;
               case 2: cls = c2; break; case 3: cls = c3; break;
               default: cls = c4; }
  const int h = LH[l], w = LW[l];
  const int plane = h * w;
  const int n = 3 * plane;
  const int k = LK[l];
  const int base = LBASE[l];
  const float* clsB = cls + (size_t)b * 3 * plane;   // [A,h,w] for this image
  const int tid = threadIdx.x, nt = blockDim.x;
  const size_t outBase = (size_t)b * KTOT + LKOFF[l];

  // objectness index i = (y*w+x)*3 + a  ->  clsB[a*plane + (y*w+x)]
  if (k >= n) {                                      // level 4: take everything
    for (int i = tid; i < n; i += nt) {
      int a = i % 3, hw = i / 3;
      selIdx[outBase + i] = base + i;
      selScr[outBase + i] = clsB[(size_t)a * plane + hw];
    }
    return;
  }

  __shared__ int hist[256];
  __shared__ unsigned sh_prefix;
  __shared__ int sh_kk, cnt_gt, cnt_eq;

  if (tid == 0) { sh_prefix = 0u; sh_kk = k; }
  __syncthreads();

  for (int shift = 24; shift >= 0; shift -= 8) {
    for (int i2 = tid; i2 < 256; i2 += nt) hist[i2] = 0;
    __syncthreads();
    const unsigned prefix = sh_prefix;
    const unsigned maskHi = (shift == 24) ? 0u : (0xFFFFFFFFu << (shift + 8));
    for (int i = tid; i < n; i += nt) {
      int a = i % 3, hw = i / 3;
      unsigned key = fkey(clsB[(size_t)a * plane + hw]);
      if ((key & maskHi) == (prefix & maskHi))
        atomicAdd(&hist[(key >> shift) & 255u], 1);
    }
    __syncthreads();
    if (tid == 0) {
      int kk = sh_kk, cum = 0, bin = 255;
      for (; bin >= 0; --bin) { cum += hist[bin]; if (cum >= kk) break; }
      if (bin < 0) bin = 0;
      sh_kk = kk - (cum - hist[bin]);                 // still needed in this bin
      sh_prefix = prefix | ((unsigned)bin << shift);
    }
    __syncthreads();
  }

  const unsigned thresh = sh_prefix;                 // k-th largest key
  const int needEq = sh_kk;
  const int baseEq = k - needEq;                     // #keys strictly greater
  if (tid == 0) { cnt_gt = 0; cnt_eq = 0; }
  __syncthreads();
  for (int i = tid; i < n; i += nt) {
    int a = i % 3, hw = i / 3;
    float f = clsB[(size_t)a * plane + hw];
    unsigned key = fkey(f);
    int slot = -1;
    if (key > thresh) slot = atomicAdd(&cnt_gt, 1);
    else if (key == thresh) {
      int e = atomicAdd(&cnt_eq, 1);
      if (e < needEq) slot = baseEq + e;
    }
    if (slot >= 0) {
      selIdx[outBase + slot] = base + i;
      selScr[outBase + slot] = f;
    }
  }
}

// ---------------- kernel 2: lazy decode of selected candidates ---------
__global__ __launch_bounds__(256) void rpn_decode(
    const float* __restrict__ x0, const float* __restrict__ x1,
    const float* __restrict__ x2, const float* __restrict__ x3,
    const float* __restrict__ x4, const float* __restrict__ anchors,
    const int* __restrict__ selIdx,
    float* __restrict__ selScr,          // in: logit, out: prob (or NEGV)
    float* __restrict__ selBox) {
  int g = blockIdx.x * blockDim.x + threadIdx.x;
  if (g >= NBATCH * KTOT) return;
  const int b = g / KTOT;
  const int gi = selIdx[g];
  const float logit = selScr[g];

  int l = 0;
#pragma unroll
  for (int t = 1; t < 5; ++t) if (gi >= LBASE[t]) l = t;
  const int li = gi - LBASE[l];
  const int h = LH[l], w = LW[l];
  const int hw = li / 3, a = li - 3 * hw;
  const float* bx;
  switch (l) { case 0: bx = x0; break; case 1: bx = x1; break;
               case 2: bx = x2; break; case 3: bx = x3; break;
               default: bx = x4; }
  const size_t plane = (size_t)h * w;
  const size_t pb = ((size_t)b * 12 + a * 4) * plane + hw;   // channel a*4+c
  float dx = bx[pb],           dy = bx[pb + plane];
  float dw = bx[pb + 2*plane], dh = bx[pb + 3*plane];

  const float* an = anchors + (size_t)gi * 4;
  float ax0 = an[0], ay0 = an[1], ax1 = an[2], ay1 = an[3];
  float aw = ax1 - ax0, ah = ay1 - ay0;
  float acx = ax0 + 0.5f * aw, acy = ay0 + 0.5f * ah;

  dw = fminf(dw, BBOX_CLIPV);
  dh = fminf(dh, BBOX_CLIPV);
  float px = dx * aw + acx, py = dy * ah + acy;
  float pw = expf(dw) * aw,  ph = expf(dh) * ah;
  float bx0 = px - 0.5f * pw, by0 = py - 0.5f * ph;
  float bx1 = px + 0.5f * pw, by1 = py + 0.5f * ph;
  bx0 = fminf(fmaxf(bx0, 0.0f), IMGSZ);
  by0 = fminf(fmaxf(by0, 0.0f), IMGSZ);
  bx1 = fminf(fmaxf(bx1, 0.0f), IMGSZ);
  by1 = fminf(fmaxf(by1, 0.0f), IMGSZ);

  float prob;
  if (logit <= -1e8f) prob = NEGV;                    // unfilled sentinel
  else {
    prob = 1.0f / (1.0f + expf(-logit));
    if (!((bx1 - bx0) >= 1.0f && (by1 - by0) >= 1.0f)) prob = NEGV;
  }
  float* o = selBox + (size_t)g * 4;
  o[0] = bx0; o[1] = by0; o[2] = bx1; o[3] = by1;
  selScr[g] = prob;
}

// ---------------- kernel 3: iterative hard NMS (one block per image) ----
__global__ __launch_bounds__(256) void rpn_nms(
    const float* __restrict__ selBox, const float* __restrict__ selScr,
    float* __restrict__ out) {
  const int b = blockIdx.x;
  const int tid = threadIdx.x;
  __shared__ float sB[KTOT * 4];     // 76288 B  (CDNA5: 320KB LDS/WGP)
  __shared__ float sS[KTOT];         // 19072 B
  __shared__ float redS[8];
  __shared__ int   redJ[8];
  __shared__ float bcS;
  __shared__ int   bcJ;

  // Stage candidates HBM -> LDS with the Tensor Data Mover (wave 0 issues,
  // TENSORcnt tracks completion), then sync the workgroup.
  if (tid < 32) {
    tdm_load_1d((unsigned)(uintptr_t)&sB[0],
                selBox + (size_t)b * KTOT * 4, (unsigned)(KTOT * 4));
    tdm_load_1d((unsigned)(uintptr_t)&sS[0],
                selScr + (size_t)b * KTOT, (unsigned)KTOT);
    tdm_wait0();
  }
  __syncthreads();

  const int CH = (KTOT + 255) / 256;       // 19 strided elements per thread
  float* outBox = out + (size_t)b * 1000 * 4;
  float* outScr = out + (size_t)NBATCH * 1000 * 4 + (size_t)b * 1000;

  for (int it = 0; it < 1000; ++it) {
    // --- argmax over LDS scores (first-occurrence tie-break: min index) ---
    float bs = -3e38f; int bj = 0;
#pragma unroll
    for (int c = 0; c < CH; ++c) {
      int j = tid + c * 256;
      if (j < KTOT) {
        float s = sS[j];
        if (s > bs) { bs = s; bj = j; }
      }
    }
#pragma unroll
    for (int off = 16; off > 0; off >>= 1) {         // wave32 reduction
      float os = __shfl_down(bs, off, 32);
      int   oj = __shfl_down(bj, off, 32);
      if (os > bs || (os == bs && oj < bj)) { bs = os; bj = oj; }
    }
    if ((tid & 31) == 0) { redS[tid >> 5] = bs; redJ[tid >> 5] = bj; }
    __syncthreads();
    if (tid == 0) {
      float rs = redS[0]; int rj = redJ[0];
#pragma unroll
      for (int i2 = 1; i2 < 8; ++i2)
        if (redS[i2] > rs || (redS[i2] == rs && redJ[i2] < rj)) {
          rs = redS[i2]; rj = redJ[i2];
        }
      bcS = rs; bcJ = rj;
      const bool valid = rs > (NEGV * 0.5f);
      float o0 = 0.f, o1 = 0.f, o2 = 0.f, o3 = 0.f;
      if (valid) { o0 = sB[rj*4+0]; o1 = sB[rj*4+1];
                   o2 = sB[rj*4+2]; o3 = sB[rj*4+3]; }
      outBox[(size_t)it * 4 + 0] = o0;
      outBox[(size_t)it * 4 + 1] = o1;
      outBox[(size_t)it * 4 + 2] = o2;
      outBox[(size_t)it * 4 + 3] = o3;
      outScr[it] = valid ? rs : NEGV;
      sS[rj] = NEGV;                                  // remove chosen
    }
    __syncthreads();

    // --- suppress everything with IoU > 0.7 vs. the chosen box ---
    const int j0 = bcJ;
    const float q0 = sB[j0*4+0], q1 = sB[j0*4+1];
    const float q2 = sB[j0*4+2], q3 = sB[j0*4+3];
    const float qa = (q2 - q0) * (q3 - q1);
#pragma unroll
    for (int c = 0; c < CH; ++c) {
      int j = tid + c * 256;
      if (j < KTOT) {
        float p0 = sB[j*4+0], p1 = sB[j*4+1], p2 = sB[j*4+2], p3 = sB[j*4+3];
        float lx = fmaxf(q0, p0), ly = fmaxf(q1, p1);
        float rx = fminf(q2, p2), ry = fminf(q3, p3);
        float iw = fmaxf(rx - lx, 0.0f), ih = fmaxf(ry - ly, 0.0f);
        float inter = iw * ih;
        float pa = (p2 - p0) * (p3 - p1);
        float iou = inter / (qa + pa - inter + 1e-9f);
        if (iou > NMS_THR) sS[j] = NEGV;
      }
    }
    __syncthreads();
  }
  (void)bcS;
}

// ---------------- host-side launch ----------------
extern "C" void kernel_launch(void* const* d_in, const int* in_sizes, int n_in,
                              void* d_out, int out_size, void* d_ws,
                              size_t ws_size, hipStream_t stream) {
  // setup_inputs() dict order: cls0, box0, cls1, box1, ..., cls4, box4, anchors
  const float* c0 = (const float*)d_in[0];
  const float* b0 = (const float*)d_in[1];
  const float* c1 = (const float*)d_in[2];
  const float* b1 = (const float*)d_in[3];
  const float* c2 = (const float*)d_in[4];
  const float* b2 = (const float*)d_in[5];
  const float* c3 = (const float*)d_in[6];
  const float* b3 = (const float*)d_in[7];
  const float* c4 = (const float*)d_in[8];
  const float* b4 = (const float*)d_in[9];
  const float* anchors = (const float*)d_in[10];
  float* out = (float*)d_out;

  // workspace layout (1.83 MB total)
  int*   selIdx = (int*)d_ws;                                // B*KTOT ints
  float* selScr = (float*)((char*)d_ws + 305152);            // B*KTOT floats
  float* selBox = (float*)((char*)d_ws + 610304);            // B*KTOT*4 floats

  const int tot = NBATCH * KTOT;
  rpn_init<<<(tot + 255) / 256, 256, 0, stream>>>(selIdx, selScr);
  rpn_topk<<<NBATCH * 5, 1024, 0, stream>>>(c0, c1, c2, c3, c4, selIdx, selScr);
  rpn_decode<<<(tot + 255) / 256, 256, 0, stream>>>(b0, b1, b2, b3, b4, anchors,
                                                    selIdx, selScr, selBox);
  rpn_nms<<<NBATCH, 256, 0, stream>>>(selBox, selScr, out);

  (void)in_sizes; (void)n_in; (void)out_size; (void)ws_size;
}